// DiffDispatchLP_9096740733542
// MI455X (gfx1250) — compile-verified
//
#include <hip/hip_runtime.h>

typedef __attribute__((ext_vector_type(2))) float v2f;
typedef __attribute__((ext_vector_type(8))) float v8f;

#define T_     96
#define NVAR   480        // [c,d,yc,yd,soc] * 96
#define M_EQN  97
#define M_TOT  2480       // 97 eq + 2383 ineq = 155*16
#define B_TOT  256
#define BT     16         // batch columns per block (WMMA N)
#define NBLK   (B_TOT / BT)
#define ITERS  400

#define ETA_DT 0.23848480035423641f   // sqrt(0.91)*0.25
#define DT_ETA 0.26207152337059383f   // 0.25/sqrt(0.91)

__device__ __forceinline__ v8f wmma_f32(v2f a, v2f b, v8f c) {
  // V_WMMA_F32_16X16X4_F32 : D(16x16 f32) = A(16x4 f32) * B(4x16 f32) + C
  return __builtin_amdgcn_wmma_f32_16x16x4_f32(
      /*neg_a=*/false, a, /*neg_b=*/false, b,
      /*c_mod=*/(short)0, c, /*reuse_a=*/false, /*reuse_b=*/false);
}

// ---------------- build K (dense), K^T, kb -------------------------------

__global__ void zero_kernel(float* p, int n) {
  int i = blockIdx.x * blockDim.x + threadIdx.x;
  int stride = gridDim.x * blockDim.x;
  for (; i < n; i += stride) p[i] = 0.0f;
}

__device__ __forceinline__ void putKT(float* K, float* KT, int r, int c, float v) {
  K[r * NVAR + c] = v;
  KT[c * M_TOT + r] = v;
}

__global__ void build_kernel(float* __restrict__ K, float* __restrict__ KT,
                             float* __restrict__ kb) {
  int r = blockIdx.x * blockDim.x + threadIdx.x;
  if (r >= M_TOT) return;
  float rhs = 0.0f;
  if (r < M_EQN) {
    if (r < 96) {
      int t = r;
      putKT(K, KT, r, 384 + t, 1.0f);
      if (t > 0) putKT(K, KT, r, 384 + t - 1, -1.0f);
      putKT(K, KT, r, t, -ETA_DT);
      putKT(K, KT, r, 96 + t, DT_ETA);
    } else {
      putKT(K, KT, r, 384 + 95, 1.0f);
    }
    rhs = 0.0f;
  } else {
    int g = r - M_EQN;
    if (g < 1248) {                       // 13 per-t box/logic rows
      int t = g / 13, j = g % 13;
      switch (j) {
        case 0:  putKT(K, KT, r, t, 1.0f);        rhs = 195.0f; break;
        case 1:  putKT(K, KT, r, 96 + t, 1.0f);   rhs = 195.0f; break;
        case 2:  putKT(K, KT, r, t, -1.0f);       rhs = 0.0f;   break;
        case 3:  putKT(K, KT, r, 96 + t, -1.0f);  rhs = 0.0f;   break;
        case 4:  putKT(K, KT, r, 192 + t, -1.0f); rhs = 0.0f;   break;
        case 5:  putKT(K, KT, r, 192 + t, 1.0f);  rhs = 1.0f;   break;
        case 6:  putKT(K, KT, r, 288 + t, -1.0f); rhs = 0.0f;   break;
        case 7:  putKT(K, KT, r, 288 + t, 1.0f);  rhs = 1.0f;   break;
        case 8:  putKT(K, KT, r, 384 + t, -1.0f); rhs = 0.0f;   break;
        case 9:  putKT(K, KT, r, 384 + t, 1.0f);  rhs = 800.0f; break;
        case 10: putKT(K, KT, r, 192 + t, 1.0f);
                 putKT(K, KT, r, 288 + t, 1.0f);  rhs = 1.0f;   break;
        case 11: putKT(K, KT, r, t, 1.0f);
                 putKT(K, KT, r, 192 + t, -195.0f); rhs = 0.0f; break;
        default: putKT(K, KT, r, 96 + t, 1.0f);
                 putKT(K, KT, r, 288 + t, -195.0f); rhs = 0.0f; break;
      }
    } else if (g < 1628) {                // ramp limits
      int u = g - 1248, t = u / 4 + 1, j = u % 4;
      rhs = 65.0f;
      if (j == 0)      { putKT(K, KT, r, t, 1.0f);          putKT(K, KT, r, t - 1, -1.0f); }
      else if (j == 1) { putKT(K, KT, r, t - 1, 1.0f);      putKT(K, KT, r, t, -1.0f); }
      else if (j == 2) { putKT(K, KT, r, 96 + t, 1.0f);     putKT(K, KT, r, 96 + t - 1, -1.0f); }
      else             { putKT(K, KT, r, 96 + t - 1, 1.0f); putKT(K, KT, r, 96 + t, -1.0f); }
    } else if (g < 1818) {                // switching exclusion
      int u = g - 1628, t = u / 2, j = u % 2;
      rhs = 1.0f;
      if (j == 0) { putKT(K, KT, r, 192 + t, 1.0f); putKT(K, KT, r, 288 + t + 1, 1.0f); }
      else        { putKT(K, KT, r, 288 + t, 1.0f); putKT(K, KT, r, 192 + t + 1, 1.0f); }
    } else if (g < 2382) {                // min-duration (yc block then yd block)
      int u = g - 1818;
      int base = (u < 282) ? 192 : 288;
      int w = u % 282;
      int t, k;
      if (w < 279)      { t = w / 3; k = w % 3 + 1; }
      else if (w < 281) { t = 93;    k = w - 278; }
      else              { t = 94;    k = 1; }
      putKT(K, KT, r, base + t + k, -1.0f);
      putKT(K, KT, r, base + t, 1.0f);
      if (t > 0) putKT(K, KT, r, base + t - 1, -1.0f);
      rhs = 0.0f;
    } else {                              // total charged energy
      for (int t = 0; t < 96; ++t) putKT(K, KT, r, t, 0.25f);
      rhs = 1200.0f;
    }
  }
  kb[r] = rhs;
}

// ---------------- power iteration for ||K||_2 -> tau = sig ---------------

__global__ void norm_kernel(const float* __restrict__ K,
                            const float* __restrict__ KT,
                            float* __restrict__ tau) {
  __shared__ float z[NVAR], z2[NVAR], w[M_TOT], red[256];
  int tid = threadIdx.x;
  for (int i = tid; i < NVAR; i += 256) z[i] = 1.0f;
  __syncthreads();
  float sig2 = 1.0f;
  for (int it = 0; it < 64; ++it) {
    for (int r = tid; r < M_TOT; r += 256) {
      const float* kr = K + r * NVAR;
      float s = 0.0f;
      for (int c = 0; c < NVAR; ++c) s += kr[c] * z[c];
      w[r] = s;
    }
    __syncthreads();
    float part = 0.0f;
    for (int c = tid; c < NVAR; c += 256) {
      const float* kt = KT + c * M_TOT;
      float s = 0.0f;
      for (int r = 0; r < M_TOT; ++r) s += kt[r] * w[r];
      z2[c] = s;
      part += s * s;
    }
    red[tid] = part;
    __syncthreads();
    for (int off = 128; off > 0; off >>= 1) {
      if (tid < off) red[tid] += red[tid + off];
      __syncthreads();
    }
    float nn = sqrtf(red[0]);   // ||K^T K z|| -> sigma_max^2
    sig2 = nn;
    float inv = 1.0f / nn;
    for (int c = tid; c < NVAR; c += 256) z[c] = z2[c] * inv;
    __syncthreads();
  }
  if (tid == 0) tau[0] = 0.9f * rsqrtf(sig2);   // 0.9 / sigma_max
}

// ---------------- PDHG main loop (WMMA f32) ------------------------------

// operand loads: A from row-major K/KT (8B contiguous per lane),
// B from pair-interleaved LDS: addr(r,n) = (r>>1)*32 + n*2 + (r&1)
#define LA(kk) (*reinterpret_cast<const v2f*>(aptr + (kk)))
#define LB(kk) (*reinterpret_cast<const v2f*>(bbase + (size_t)(kk) * 16))

// one "group" = 40 K-values = 5 chunks of 8 = 10 WMMAs
#define LOADG(AA, BB, kbase)                                  \
  _Pragma("unroll")                                           \
  for (int c5 = 0; c5 < 5; ++c5) {                            \
    AA[c5][0] = LA((kbase) + 8 * c5);                         \
    AA[c5][1] = LA((kbase) + 8 * c5 + 4);                     \
    BB[c5][0] = LB((kbase) + 8 * c5);                         \
    BB[c5][1] = LB((kbase) + 8 * c5 + 4);                     \
  }
#define COMPG(AA, BB)                                         \
  _Pragma("unroll")                                           \
  for (int c5 = 0; c5 < 5; ++c5) {                            \
    acc0 = wmma_f32(AA[c5][0], BB[c5][0], acc0);              \
    acc1 = wmma_f32(AA[c5][1], BB[c5][1], acc1);              \
  }

__global__ void __launch_bounds__(256) pdhg_kernel(
    const float* __restrict__ price, const float* __restrict__ K,
    const float* __restrict__ KT, const float* __restrict__ kb,
    const float* __restrict__ taup, float* __restrict__ out) {
  __shared__ float Ylds[M_TOT * BT];    // 158,720 B  (pair-interleaved)
  __shared__ float Xbar[NVAR * BT];     //  30,720 B  (pair-interleaved)
  __shared__ float Qlds[NVAR * BT];     //  30,720 B  (plain r*16+n)

  const int tid  = threadIdx.x;
  const int lane = tid & 31;
  const int wave = tid >> 5;        // 8 waves
  const int n    = lane & 15;       // batch column / WMMA M,N index
  const int half = lane >> 4;       // 0/1 lane-half (K offset in A/B layouts)
  const int b    = blockIdx.x * BT + n;

  const float tau = taup[0];
  const float sig = tau;

  for (int i = tid; i < M_TOT * BT; i += 256) Ylds[i] = 0.0f;

  // X state in registers; wave owns n-tiles {wave, wave+8, wave+16, wave+24}
  float xreg[4][8];
#pragma unroll
  for (int j = 0; j < 4; ++j) {
    int nt = wave + 8 * j;
#pragma unroll
    for (int v = 0; v < 8; ++v) {
      xreg[j][v] = 0.0f;
      if (nt < 30) {
        int rrow = nt * 16 + v + half * 8;
        float q = 0.0f;
        if (rrow < 96)       q =  0.25f * price[b * T_ + rrow];
        else if (rrow < 192) q = -0.25f * price[b * T_ + (rrow - 96)];
        Qlds[rrow * 16 + n] = q;
      }
    }
  }
  __syncthreads();

  for (int it = 0; it < ITERS; ++it) {
    // ---- X update: x -= tau*(q + K^T y); Xbar = 2x_new - x_old ----
#pragma unroll
    for (int j = 0; j < 4; ++j) {
      int nt = wave + 8 * j;
      if (nt < 30) {                        // wave-uniform guard
        v8f acc0 = {}, acc1 = {};
        const float* aptr  = KT + (size_t)(nt * 16 + n) * M_TOT + half * 2;
        const float* bbase = &Ylds[half * 32 + n * 2];
        v2f A0[5][2], B0[5][2], A1[5][2], B1[5][2];
        // ping-pong prefetch: group g+1 in flight while computing group g
        LOADG(A0, B0, 0);
#pragma unroll 1
        for (int kg = 0; kg + 80 < M_TOT; kg += 80) {
          LOADG(A1, B1, kg + 40);
          COMPG(A0, B0);
          LOADG(A0, B0, kg + 80);
          COMPG(A1, B1);
        }
        LOADG(A1, B1, M_TOT - 40);
        COMPG(A0, B0);
        COMPG(A1, B1);
#pragma unroll
        for (int v = 0; v < 8; ++v) {
          int rrow = nt * 16 + v + half * 8;
          float xo = xreg[j][v];
          float xn = xo - tau * (Qlds[rrow * 16 + n] + acc0[v] + acc1[v]);
          xreg[j][v] = xn;
          Xbar[(rrow >> 1) * 32 + n * 2 + (rrow & 1)] = 2.0f * xn - xo;
        }
      }
    }
    __syncthreads();

    // ---- Y update: y = proj(y + sig*(K xbar - kb)) ----
    for (int rt = wave; rt < 155; rt += 8) {
      v8f acc0 = {}, acc1 = {};
      const float* aptr  = K + (size_t)(rt * 16 + n) * NVAR + half * 2;
      const float* bbase = &Xbar[half * 32 + n * 2];
      v2f A0[5][2], B0[5][2], A1[5][2], B1[5][2];
      LOADG(A0, B0, 0);
#pragma unroll 1
      for (int kg = 0; kg + 80 < NVAR; kg += 80) {
        LOADG(A1, B1, kg + 40);
        COMPG(A0, B0);
        LOADG(A0, B0, kg + 80);
        COMPG(A1, B1);
      }
      LOADG(A1, B1, NVAR - 40);
      COMPG(A0, B0);
      COMPG(A1, B1);
#pragma unroll
      for (int v = 0; v < 8; ++v) {
        int rrow = rt * 16 + v + half * 8;
        float* yp = &Ylds[(rrow >> 1) * 32 + n * 2 + (rrow & 1)];
        float yn = *yp + sig * (acc0[v] + acc1[v] - kb[rrow]);
        if (rrow >= M_EQN) yn = fmaxf(yn, 0.0f);
        *yp = yn;
      }
    }
    __syncthreads();
  }

  // ---- write c = x[0:96], d = x[96:192] ----
#pragma unroll
  for (int j = 0; j < 4; ++j) {
    int nt = wave + 8 * j;
    if (nt < 12) {
#pragma unroll
      for (int v = 0; v < 8; ++v) {
        int rrow = nt * 16 + v + half * 8;
        int sel = rrow / 96;
        int t = rrow % 96;
        out[sel * (B_TOT * T_) + b * T_ + t] = xreg[j][v];
      }
    }
  }
}

// ---------------- host launch --------------------------------------------

extern "C" void kernel_launch(void* const* d_in, const int* in_sizes, int n_in,
                              void* d_out, int out_size, void* d_ws, size_t ws_size,
                              hipStream_t stream) {
  (void)in_sizes; (void)n_in; (void)out_size; (void)ws_size;
  const float* price = (const float*)d_in[0];
  float* out = (float*)d_out;
  float* ws  = (float*)d_ws;

  float* K   = ws;                        // 2480*480
  float* KT  = K  + M_TOT * NVAR;         // 480*2480
  float* kb  = KT + M_TOT * NVAR;         // 2480
  float* tau = kb + M_TOT;                // 1 (+pad)

  zero_kernel<<<1024, 256, 0, stream>>>(K, 2 * M_TOT * NVAR);
  build_kernel<<<(M_TOT + 255) / 256, 256, 0, stream>>>(K, KT, kb);
  norm_kernel<<<1, 256, 0, stream>>>(K, KT, tau);
  pdhg_kernel<<<NBLK, 256, 0, stream>>>(price, K, KT, kb, tau, out);
}